// HungarianMatcher_21672404976057
// MI455X (gfx1250) — compile-verified
//
#include <hip/hip_runtime.h>

// Hungarian matcher cost-matrix, CDNA5 (gfx1250), wave32.
// class + quadratic (viz/center) part via V_WMMA_F32_16X16X4_F32 (rank-21 GEMM,
// padded to K=24 -> 6 chained WMMAs per 16x16 tile); weighted-L1 (offset/abs)
// part in VALU on the same tile using the WMMA C/D lane layout.
// Loop order kp-outer / row-inner + launch_bounds(128,4) keeps VGPRs < 256
// (no s_set_vgpr_msb thrash, 4 waves/SIMD occupancy).

typedef __attribute__((ext_vector_type(2))) float v2f;
typedef __attribute__((ext_vector_type(8))) float v8f;

#define BSZ   8
#define QN    500
#define NP    (BSZ * QN)   // 4000 preds
#define TN    400          // targets
#define KF    24           // padded feature length (21 -> 24)
#define JT    (TN / 16)    // 25 target tiles
#define NWAVE 4

__global__ __launch_bounds__(128, 4) void hmatch_cost_kernel(
    const float* __restrict__ pred_logits,   // [NP,2]
    const float* __restrict__ pred_kp,       // [NP,53]
    const int*   __restrict__ tgt_ids,       // [TN]
    const float* __restrict__ tgt_kp,        // [TN,53]
    const int*   __restrict__ num_boxes,     // [1]
    float* __restrict__ out)                 // [NP,TN]
{
    __shared__ float Bfeat[TN][KF];   // target WMMA features (Vg|Cg|onehot)  38400 B
    __shared__ float Gconst[TN];      // per-target quad constant              1600 B
    __shared__ float Pfeat[16][KF];   // pred WMMA features                    1536 B
    __shared__ float Pconst[16];      // per-pred quad constant                  64 B
    __shared__ float PZp[16][34];     // pred Z deltas                         2176 B
    __shared__ float PCp[16][2];      // pred centers                           128 B

    const int tid   = threadIdx.x;
    const int itile = blockIdx.x;               // 0..249
    const float nb     = (float)num_boxes[0];
    const float inv_nb = 1.0f / nb;

    // ---- stage target features (all 400 rows, cooperatively) ----
    for (int t = tid; t < TN; t += 128) {
        const float* tk = tgt_kp + t * 53;
        float cg0 = tk[0], cg1 = tk[1];
        float sv = 0.0f;
        #pragma unroll
        for (int k = 0; k < 17; ++k) {
            float v = tk[36 + k];
            Bfeat[t][k] = v;
            sv += v * v;
        }
        Bfeat[t][17] = cg0;
        Bfeat[t][18] = cg1;
        int id = tgt_ids[t];
        Bfeat[t][19] = (id == 0) ? 1.0f : 0.0f;
        Bfeat[t][20] = (id == 1) ? 1.0f : 0.0f;
        Bfeat[t][21] = 0.0f; Bfeat[t][22] = 0.0f; Bfeat[t][23] = 0.0f;
        Gconst[t] = (0.2f * sv + 0.5f * (cg0 * cg0 + cg1 * cg1)) * inv_nb;
    }

    // ---- stage pred features for this 16-row tile ----
    if (tid < 16) {
        const int ip = itile * 16 + tid;
        float l0 = pred_logits[ip * 2 + 0];
        float l1 = pred_logits[ip * 2 + 1];
        float mx = fmaxf(l0, l1);
        float e0 = expf(l0 - mx), e1 = expf(l1 - mx);
        float inv_s = 1.0f / (e0 + e1);
        float p0 = e0 * inv_s, p1 = e1 * inv_s;

        const float* pk = pred_kp + ip * 53;
        float cp0 = pk[0], cp1 = pk[1];
        float sv = 0.0f;
        #pragma unroll
        for (int k = 0; k < 17; ++k) {
            float v = pk[36 + k];
            Pfeat[tid][k] = (-0.4f * inv_nb) * v;
            sv += v * v;
        }
        Pfeat[tid][17] = -inv_nb * cp0;
        Pfeat[tid][18] = -inv_nb * cp1;
        Pfeat[tid][19] = -p0;
        Pfeat[tid][20] = -p1;
        Pfeat[tid][21] = 0.0f; Pfeat[tid][22] = 0.0f; Pfeat[tid][23] = 0.0f;
        Pconst[tid] = (0.2f * sv + 0.5f * (cp0 * cp0 + cp1 * cp1)) * inv_nb;
        PCp[tid][0] = cp0;
        PCp[tid][1] = cp1;
        #pragma unroll
        for (int k = 0; k < 34; ++k) PZp[tid][k] = pk[2 + k];
    }
    __syncthreads();

    const int wave  = tid >> 5;
    const int lane  = tid & 31;
    const int hi    = lane >> 4;     // lane half selects K pair (A/B) and M+8 (C/D)
    const int row   = lane & 15;     // A row / B column / C column
    const int khalf = 2 * hi;
    const int mbase = 8 * hi;        // C/D rows handled by this lane: mbase..mbase+7

    // A fragments (invariant across target tiles): lane holds K = 4s+khalf, +1
    v2f afrag[6];
    #pragma unroll
    for (int s = 0; s < 6; ++s) {
        afrag[s][0] = Pfeat[row][4 * s + khalf];
        afrag[s][1] = Pfeat[row][4 * s + khalf + 1];
    }

    for (int jt = wave; jt < JT; jt += NWAVE) {
        const int jbase = jt * 16;
        const int j     = jbase + row;   // this lane's output column

        // prefetch next tile's target row (CDNA5 global_prefetch_b8)
        {
            int jn = jbase + NWAVE * 16;
            if (jn < TN) __builtin_prefetch(tgt_kp + (jn + row) * 53, 0, 0);
        }

        // ---- class + quadratic part: 16x16 tile via f32 WMMA, K = 24 ----
        v8f c = {};
        #pragma unroll
        for (int s = 0; s < 6; ++s) {
            v2f b;
            b[0] = Bfeat[j][4 * s + khalf];
            b[1] = Bfeat[j][4 * s + khalf + 1];
            c = __builtin_amdgcn_wmma_f32_16x16x4_f32(
                    /*neg_a=*/false, afrag[s],
                    /*neg_b=*/false, b,
                    /*c_mod=*/(short)0, c,
                    /*reuse_a=*/false, /*reuse_b=*/false);
        }

        // ---- weighted-L1 (offset + abs) part for the same tile ----
        // target Z from global (34 regs); Vg/Cg reused from LDS Bfeat.
        const float* tk = tgt_kp + j * 53;
        float zg[34];
        #pragma unroll
        for (int k = 0; k < 34; ++k) zg[k] = tk[2 + k];
        const float cg0 = Bfeat[j][17];
        const float cg1 = Bfeat[j][18];
        const float gj  = Gconst[j];

        float dc0[8], dc1[8], aoff[8], aabs[8];
        #pragma unroll
        for (int r = 0; r < 8; ++r) {
            const int m = mbase + r;
            dc0[r] = PCp[m][0] - cg0;
            dc1[r] = PCp[m][1] - cg1;
            aoff[r] = 0.0f;
            aabs[r] = 0.0f;
        }

        #pragma unroll
        for (int kp = 0; kp < 17; ++kp) {
            const float v   = Bfeat[j][kp];     // Vg[kp] staged in LDS
            const float zg0 = zg[2 * kp];
            const float zg1 = zg[2 * kp + 1];
            #pragma unroll
            for (int r = 0; r < 8; ++r) {
                const int m = mbase + r;
                float dz0 = PZp[m][2 * kp]     - zg0;
                float dz1 = PZp[m][2 * kp + 1] - zg1;
                aoff[r] += (fabsf(dz0) + fabsf(dz1)) * v;
                aabs[r] += (fabsf(dz0 + dc0[r]) + fabsf(dz1 + dc1[r])) * v;
            }
        }

        #pragma unroll
        for (int r = 0; r < 8; ++r) {
            const int m = mbase + r;
            float val = c[r] + Pconst[m] + gj
                      + (0.5f * aoff[r] + 4.0f * aabs[r]) * inv_nb;
            out[(itile * 16 + m) * TN + j] = val;
        }
    }
}

extern "C" void kernel_launch(void* const* d_in, const int* in_sizes, int n_in,
                              void* d_out, int out_size, void* d_ws, size_t ws_size,
                              hipStream_t stream) {
    (void)in_sizes; (void)n_in; (void)d_ws; (void)ws_size; (void)out_size;
    const float* pred_logits = (const float*)d_in[0];
    const float* pred_kp     = (const float*)d_in[1];
    const int*   tgt_ids     = (const int*)d_in[2];
    const float* tgt_kp      = (const float*)d_in[3];
    const int*   num_boxes   = (const int*)d_in[4];
    float* out = (float*)d_out;

    dim3 grid(NP / 16);   // 250 tiles of 16 preds
    dim3 block(128);      // 4 waves; each wave owns full 16x16 WMMA tiles
    hmatch_cost_kernel<<<grid, block, 0, stream>>>(
        pred_logits, pred_kp, tgt_ids, tgt_kp, num_boxes, out);
}